// Encoder_13615046328411
// MI455X (gfx1250) — compile-verified
//
#include <hip/hip_runtime.h>
#include <math.h>

// ---------------- problem constants ----------------
#define BB   2
#define SS   2048
#define DD   512
#define HH   8
#define HDIM 64
#define FFN  2048
#define LAYERS 2
#define MROWS (BB * SS)          // 4096
#define NEGV (-1.0e9f)

// ---------------- WMMA helpers (f32 16x16x4) ----------------
typedef __attribute__((ext_vector_type(2))) float v2f;
typedef __attribute__((ext_vector_type(8))) float v8f;

__device__ __forceinline__ v8f wmma4(v2f a, v2f b, v8f c) {
  // D = A(16x4) * B(4x16) + C(16x16), all f32, wave32
  return __builtin_amdgcn_wmma_f32_16x16x4_f32(false, a, false, b, (short)0, c,
                                               false, false);
}

__device__ __forceinline__ v8f zero8() {
  v8f z = {0.f, 0.f, 0.f, 0.f, 0.f, 0.f, 0.f, 0.f};
  return z;
}

__device__ __forceinline__ float half_max16(float v) {
  // max across the 16 lanes of each half-wave (xor masks stay in-half, wave32)
  v = fmaxf(v, __shfl_xor(v, 1));
  v = fmaxf(v, __shfl_xor(v, 2));
  v = fmaxf(v, __shfl_xor(v, 4));
  v = fmaxf(v, __shfl_xor(v, 8));
  return v;
}
__device__ __forceinline__ float half_sum16(float v) {
  v += __shfl_xor(v, 1);
  v += __shfl_xor(v, 2);
  v += __shfl_xor(v, 4);
  v += __shfl_xor(v, 8);
  return v;
}

// ---------------- GEMM: C[M,N] = A[M,K] @ W[K,N] (+bias)(+relu) ----------------
// block tile 128x64, 8 waves (4 in M x 2 in N), wave tile 32x32 (4 accums).
// LDS tiles stored K-pair-packed as float2 so every WMMA fragment is one
// aligned ds_load_b64 (no repacking movs).
#define KS 32
__global__ __launch_bounds__(256) void gemm_kernel(
    const float* __restrict__ A, const float* __restrict__ W,
    const float* __restrict__ bias, float* __restrict__ C, int M, int K, int N,
    int relu) {
  __shared__ v2f Ap[KS / 2][128 + 1];  // Ap[k2][m] = {A[m][2k2], A[m][2k2+1]}
  __shared__ v2f Bp[KS / 2][64 + 1];   // Bp[k2][n] = {B[2k2][n], B[2k2+1][n]}

  const int tid = threadIdx.x;
  const int lane = tid & 31;
  const int wave = tid >> 5;
  const int waveM = wave & 3;   // 0..3 -> 32 rows each
  const int waveN = wave >> 2;  // 0..1 -> 32 cols each
  const int ln = lane & 15;
  const int kh = lane >> 4;
  const int m0 = blockIdx.y * 128;
  const int n0 = blockIdx.x * 64;

  v8f acc00 = zero8(), acc01 = zero8(), acc10 = zero8(), acc11 = zero8();

  // A stage mapping: thread -> row (tid>>1), 16 consecutive k
  const int mA = tid >> 1;             // 0..127
  const int kA = (tid & 1) * 16;       // 0 or 16
  // B stage mapping: thread -> k-pair (tid>>4), 4 consecutive n
  const int k2B = tid >> 4;            // 0..15
  const int nB = (tid & 15) * 4;       // 0..60

  const float* ga = A + (size_t)(m0 + mA) * K + kA;
  const float* gb0 = W + (size_t)(2 * k2B) * N + n0 + nB;
  const float* gb1 = gb0 + N;

  for (int kk = 0; kk < K; kk += KS) {
#pragma unroll
    for (int j = 0; j < 8; ++j) {
      v2f t;
      t.x = ga[kk + 2 * j + 0];
      t.y = ga[kk + 2 * j + 1];
      Ap[(kA >> 1) + j][mA] = t;
    }
#pragma unroll
    for (int i = 0; i < 4; ++i) {
      v2f t;
      t.x = gb0[(size_t)kk * N + i];
      t.y = gb1[(size_t)kk * N + i];
      Bp[k2B][nB + i] = t;
    }
    if (kk + KS < K) {  // overlap next slab's L2->WGP latency
      __builtin_prefetch(ga + kk + KS, 0, 3);
      __builtin_prefetch(gb0 + (size_t)(kk + KS) * N, 0, 3);
    }
    __syncthreads();

#pragma unroll
    for (int ks = 0; ks < KS; ks += 4) {
      const int k2 = (ks >> 1) + kh;  // pair index of (ks + 2*kh)
      v2f a0 = Ap[k2][waveM * 32 + ln];
      v2f a1 = Ap[k2][waveM * 32 + 16 + ln];
      v2f b0 = Bp[k2][waveN * 32 + ln];
      v2f b1 = Bp[k2][waveN * 32 + 16 + ln];
      acc00 = wmma4(a0, b0, acc00);
      acc01 = wmma4(a0, b1, acc01);
      acc10 = wmma4(a1, b0, acc10);
      acc11 = wmma4(a1, b1, acc11);
    }
    __syncthreads();
  }

  const int c0 = n0 + waveN * 32 + ln;
  const int c1 = c0 + 16;
  const float bv0 = bias ? bias[c0] : 0.f;
  const float bv1 = bias ? bias[c1] : 0.f;
#pragma unroll
  for (int r = 0; r < 8; ++r) {
    const int rowA = m0 + waveM * 32 + r + 8 * kh;  // D layout: row = r + 8*half
    const int rowB = rowA + 16;
    float v00 = acc00[r] + bv0, v01 = acc01[r] + bv1;
    float v10 = acc10[r] + bv0, v11 = acc11[r] + bv1;
    if (relu) {
      v00 = fmaxf(v00, 0.f);
      v01 = fmaxf(v01, 0.f);
      v10 = fmaxf(v10, 0.f);
      v11 = fmaxf(v11, 0.f);
    }
    C[(size_t)rowA * N + c0] = v00;
    C[(size_t)rowA * N + c1] = v01;
    C[(size_t)rowB * N + c0] = v10;
    C[(size_t)rowB * N + c1] = v11;
  }
}

// ---------------- Flash attention over contiguous 192-float qkv rows ----------
// grid: (S/64, H, B), block: 128 (4 waves, each owns 16 q rows).
// All LDS tiles K-pair-packed float2 -> every fragment is one ds_load_b64.
__global__ __launch_bounds__(128) void attn_kernel(
    const float* __restrict__ qkv, const unsigned char* __restrict__ mask,
    float* __restrict__ out) {
  __shared__ v2f kTp[HDIM / 2][16 + 1];  // kTp[kk2][key] = {K[key][2kk2], K[key][2kk2+1]}
  __shared__ v2f vsp[8][HDIM + 1];       // vsp[kp][hd]  = {V[2kp][hd], V[2kp+1][hd]}
  __shared__ v2f psp[4][16][8 + 1];      // psp[w][m][c2] = {P[m][2c2], P[m][2c2+1]}

  const int tid = threadIdx.x;
  const int lane = tid & 31;
  const int wave = tid >> 5;  // 0..3
  const int ln = lane & 15;
  const int kh = lane >> 4;
  const int b = blockIdx.z;
  const int h = blockIdx.y;
  const int rowbase = (b * HH + h) * SS;       // qkv row (of 192 floats)
  const int q0 = blockIdx.x * 64 + wave * 16;  // this wave's q-tile

  // preload q A-fragments (K = 64 -> 16 wmma steps)
  const float* qrow = qkv + (size_t)(rowbase + q0 + ln) * 192;
  v2f qf[16];
#pragma unroll
  for (int t = 0; t < 16; ++t) {
    qf[t].x = qrow[4 * t + 2 * kh + 0];
    qf[t].y = qrow[4 * t + 2 * kh + 1];
  }

  v8f o0 = zero8(), o1 = zero8(), o2 = zero8(), o3 = zero8();
  float m_r[8], l_r[8];
#pragma unroll
  for (int r = 0; r < 8; ++r) {
    m_r[r] = -INFINITY;
    l_r[r] = 0.f;
  }

  const int keyld = tid >> 3;      // 0..15 (K staging)
  const int coff = (tid & 7) * 8;  // 0..56
  const int kpld = tid >> 4;       // 0..7  (V staging: key pair)
  const int cv = (tid & 15) * 4;   // 0..60

  for (int kt = 0; kt < SS / 16; ++kt) {
    const int kbase = kt * 16;
    // stage K (transposed, pair-packed along hd) and V (pair-packed along key)
    const float* krow = qkv + (size_t)(rowbase + kbase + keyld) * 192 + 64;
#pragma unroll
    for (int j = 0; j < 4; ++j) {
      v2f t;
      t.x = krow[coff + 2 * j + 0];
      t.y = krow[coff + 2 * j + 1];
      kTp[(coff >> 1) + j][keyld] = t;
    }
    const float* vrow0 = qkv + (size_t)(rowbase + kbase + 2 * kpld) * 192 + 128;
    const float* vrow1 = vrow0 + 192;
#pragma unroll
    for (int i = 0; i < 4; ++i) {
      v2f t;
      t.x = vrow0[cv + i];
      t.y = vrow1[cv + i];
      vsp[kpld][cv + i] = t;
    }
    __syncthreads();

    // scores: S(16x16) = Q(16x64) @ K^T(64x16)
    v8f s = zero8();
#pragma unroll
    for (int ks = 0; ks < HDIM; ks += 4) {
      v2f bf = kTp[(ks >> 1) + kh][ln];
      s = wmma4(qf[ks >> 2], bf, s);
    }

    // online softmax update; lane's column key is fixed across rows
    const float cmask = mask[b * SS + kbase + ln] ? 1.f : 0.f;
#pragma unroll
    for (int r = 0; r < 8; ++r) {
      float sv = (cmask != 0.f) ? s[r] * 0.125f : NEGV;  // 1/sqrt(64)
      float mx = half_max16(sv);
      float mnew = fmaxf(m_r[r], mx);
      float p = __expf(sv - mnew);
      float corr = __expf(m_r[r] - mnew);
      float psum = half_sum16(p);
      l_r[r] = l_r[r] * corr + psum;
      m_r[r] = mnew;
      o0[r] *= corr;
      o1[r] *= corr;
      o2[r] *= corr;
      o3[r] *= corr;
      // scatter into pair-packed P slab (this wave's slice only)
      float* pp = (float*)&psp[wave][r + 8 * kh][ln >> 1];
      pp[ln & 1] = p;
    }
    __syncthreads();

    // O(16x64) += P(16x16) @ V(16x64)
#pragma unroll
    for (int t = 0; t < 4; ++t) {  // K=16 -> 4 wmma steps
      v2f a = psp[wave][ln][2 * t + kh];
      v2f b0 = vsp[2 * t + kh][0 + ln];
      v2f b1 = vsp[2 * t + kh][16 + ln];
      v2f b2 = vsp[2 * t + kh][32 + ln];
      v2f b3 = vsp[2 * t + kh][48 + ln];
      o0 = wmma4(a, b0, o0);
      o1 = wmma4(a, b1, o1);
      o2 = wmma4(a, b2, o2);
      o3 = wmma4(a, b3, o3);
    }
    __syncthreads();
  }

  // normalize, apply row mask, store (output rows of 64 are contiguous)
#pragma unroll
  for (int r = 0; r < 8; ++r) {
    const int qg = q0 + r + 8 * kh;
    const float rm = mask[b * SS + qg] ? 1.f : 0.f;
    const float inv = (l_r[r] > 0.f) ? rm / l_r[r] : 0.f;
    float* orow = out + (size_t)(rowbase + qg) * HDIM;
    orow[0 + ln] = o0[r] * inv;
    orow[16 + ln] = o1[r] * inv;
    orow[32 + ln] = o2[r] * inv;
    orow[48 + ln] = o3[r] * inv;
  }
}

// ---------------- Add + LayerNorm: out = g*norm(x+res)+be ----------------
__global__ __launch_bounds__(256) void ln_kernel(
    const float* __restrict__ x, const float* __restrict__ res,
    const float* __restrict__ g, const float* __restrict__ be,
    float* __restrict__ out) {
  const int row = blockIdx.x;
  const int tid = threadIdx.x;
  const float* xr = x + (size_t)row * DD;
  const float* rr = res + (size_t)row * DD;
  float v0 = xr[tid] + rr[tid];
  float v1 = xr[tid + 256] + rr[tid + 256];
  float s = v0 + v1;
  float sq = v0 * v0 + v1 * v1;
#pragma unroll
  for (int msk = 1; msk < 32; msk <<= 1) {
    s += __shfl_xor(s, msk);
    sq += __shfl_xor(sq, msk);
  }
  __shared__ float ssum[8], ssq[8];
  if ((tid & 31) == 0) {
    ssum[tid >> 5] = s;
    ssq[tid >> 5] = sq;
  }
  __syncthreads();
  float ts = 0.f, tq = 0.f;
#pragma unroll
  for (int i = 0; i < 8; ++i) {
    ts += ssum[i];
    tq += ssq[i];
  }
  const float mean = ts * (1.f / DD);
  const float var = tq * (1.f / DD) - mean * mean;
  const float rstd = rsqrtf(var + 1e-5f);
  out[(size_t)row * DD + tid] = g[tid] * (v0 - mean) * rstd + be[tid];
  out[(size_t)row * DD + tid + 256] =
      g[tid + 256] * (v1 - mean) * rstd + be[tid + 256];
}

// ---------------- launch ----------------
extern "C" void kernel_launch(void* const* d_in, const int* in_sizes, int n_in,
                              void* d_out, int out_size, void* d_ws,
                              size_t ws_size, hipStream_t stream) {
  const float* x = (const float*)d_in[0];
  const unsigned char* mask = (const unsigned char*)d_in[1];  // jnp bool = 1B
  const float* Wqkv = (const float*)d_in[2];
  const float* Wo = (const float*)d_in[3];
  const float* bo = (const float*)d_in[4];
  const float* g1 = (const float*)d_in[5];
  const float* be1 = (const float*)d_in[6];
  const float* W1 = (const float*)d_in[7];
  const float* bf1 = (const float*)d_in[8];
  const float* W2 = (const float*)d_in[9];
  const float* bf2 = (const float*)d_in[10];
  const float* g2 = (const float*)d_in[11];
  const float* be2 = (const float*)d_in[12];

  float* ws = (float*)d_ws;
  // region lifetimes: qkv (24MB) and ffn-hidden (32MB) are disjoint -> share
  float* big = ws;              // 8,388,608 floats
  float* a1 = ws + 8388608;     // 2,097,152
  float* yb = ws + 10485760;    // 2,097,152 (y1 then y2)
  float* x2 = ws + 12582912;    // 2,097,152
  float* xmid = ws + 14680064;  // 2,097,152

  const float* xin = x;
  for (int i = 0; i < LAYERS; ++i) {
    const float* Wqkv_i = Wqkv + (size_t)i * DD * 3 * DD;
    const float* Wo_i = Wo + (size_t)i * DD * DD;
    const float* bo_i = bo + (size_t)i * DD;
    const float* g1_i = g1 + (size_t)i * DD;
    const float* be1_i = be1 + (size_t)i * DD;
    const float* W1_i = W1 + (size_t)i * DD * FFN;
    const float* bf1_i = bf1 + (size_t)i * FFN;
    const float* W2_i = W2 + (size_t)i * FFN * DD;
    const float* bf2_i = bf2 + (size_t)i * DD;
    const float* g2_i = g2 + (size_t)i * DD;
    const float* be2_i = be2 + (size_t)i * DD;

    float* qkv = big;
    float* f1 = big;

    // QKV = x @ Wqkv  [4096,512]x[512,1536]
    gemm_kernel<<<dim3(3 * DD / 64, MROWS / 128), 256, 0, stream>>>(
        xin, Wqkv_i, nullptr, qkv, MROWS, DD, 3 * DD, 0);
    // attention -> a1 (contiguous == [B,S,D] after the reference reshapes)
    attn_kernel<<<dim3(SS / 64, HH, BB), 128, 0, stream>>>(qkv, mask, a1);
    // y1 = a1 @ Wo + bo
    gemm_kernel<<<dim3(DD / 64, MROWS / 128), 256, 0, stream>>>(
        a1, Wo_i, bo_i, yb, MROWS, DD, DD, 0);
    // x2 = LN(y1 + xin)
    ln_kernel<<<MROWS, 256, 0, stream>>>(yb, xin, g1_i, be1_i, x2);
    // f1 = relu(x2 @ W1 + bf1)
    gemm_kernel<<<dim3(FFN / 64, MROWS / 128), 256, 0, stream>>>(
        x2, W1_i, bf1_i, f1, MROWS, DD, FFN, 1);
    // y2 = f1 @ W2 + bf2
    gemm_kernel<<<dim3(DD / 64, MROWS / 128), 256, 0, stream>>>(
        f1, W2_i, bf2_i, yb, MROWS, FFN, DD, 0);
    // xout = LN(y2 + x2)
    float* xo = (i == LAYERS - 1) ? (float*)d_out : xmid;
    ln_kernel<<<MROWS, 256, 0, stream>>>(yb, x2, g2_i, be2_i, xo);
    xin = xo;
  }
}